// DGCNNClassifier_37847251812430
// MI455X (gfx1250) — compile-verified
//
#include <hip/hip_runtime.h>

// DGCNN classifier for MI455X (gfx1250, wave32). All GEMM-shaped compute uses
// V_WMMA_F32_16X16X4_F32 (fp32 matrix path) to match reference numerics while
// exercising the CDNA5 WMMA pipes. Working set (~90MB) fits in the 192MB L2.
// Layer-1 input/weights are zero-padded (C=3 -> 4) so every WMMA K-loop is
// branch-free (clause-batched loads, staged s_wait_loadcnt before the WMMAs).
// Top-k row staging uses GLOBAL_LOAD_ASYNC_TO_LDS_B128 (ASYNCcnt) to fill LDS
// without a VGPR round-trip.

typedef __attribute__((ext_vector_type(2))) float v2f;
typedef __attribute__((ext_vector_type(8))) float v8f;

static __device__ inline v8f wmma4(v2f a, v2f b, v8f c) {
  // 8 args: (neg_a, A, neg_b, B, c_mod, C, reuse_a, reuse_b)
  return __builtin_amdgcn_wmma_f32_16x16x4_f32(false, a, false, b, (short)0, c,
                                               false, false);
}

static __device__ inline v2f ld2(const float* __restrict__ p) {
  return *(const v2f*)p;  // 8-byte aligned contiguous pair
}

// Async copy 16 bytes global -> LDS (per lane). LDS byte address = low 32 bits
// of the generic pointer. Tracked by ASYNCcnt.
static __device__ inline void async_g2lds_b128(float* lds_dst, const float* gsrc) {
  unsigned lds_off = (unsigned)(size_t)lds_dst;
  asm volatile("global_load_async_to_lds_b128 %0, %1, off"
               :
               : "v"(lds_off), "v"(gsrc)
               : "memory");
}
static __device__ inline void wait_asynccnt0() {
  asm volatile("s_wait_asynccnt 0x0" ::: "memory");
}

__device__ inline float lrelu(float y) { return y > 0.f ? y : 0.2f * y; }

// ---------------------------------------------------------------- BN folding
__global__ void k_bnprep(const float* g, const float* b, const float* m,
                         const float* v, float* scale, float* shift, int C) {
  int i = blockIdx.x * blockDim.x + threadIdx.x;
  if (i < C) {
    float s = g[i] * rsqrtf(v[i] + 1e-5f);
    scale[i] = s;
    shift[i] = b[i] - m[i] * s;
  }
}

// ------------------------------------------------ pad x (B,3,N) -> xp (B,4,N)
__global__ void k_pad(const float* __restrict__ x, float* __restrict__ xp, int N) {
  int n = blockIdx.x * blockDim.x + threadIdx.x;
  int c = blockIdx.y, b = blockIdx.z;
  if (n >= N) return;
  float v = (c < 3) ? x[((size_t)b * 3 + c) * N + n] : 0.f;
  xp[((size_t)b * 4 + c) * N + n] = v;
}

// ----------------------------------- pad W1 (64,6) -> (64,8), zeros at 3 and 7
__global__ void k_padW1(const float* __restrict__ W1, float* __restrict__ W1p) {
  int t = blockIdx.x * blockDim.x + threadIdx.x;
  if (t >= 64 * 8) return;
  int o = t >> 3, c = t & 7;
  float v = 0.f;
  if (c < 3) v = W1[o * 6 + c];
  else if (c >= 4 && c < 7) v = W1[o * 6 + 3 + (c - 4)];
  W1p[t] = v;
}

// ---------------------------------------------------------------- ||x||^2
__global__ void k_sqnorm(const float* __restrict__ x, float* __restrict__ xx,
                         int C, int N) {
  int n = blockIdx.x * blockDim.x + threadIdx.x;
  int b = blockIdx.y;
  if (n >= N) return;
  const float* xb = x + (size_t)b * C * N;
  float s = 0.f;
  for (int c = 0; c < C; c++) {
    float t = xb[c * N + n];
    s += t * t;
  }
  xx[(size_t)b * N + n] = s;
}

// ------------------------------------------------- D = xx_i + xx_j - 2*x_i.x_j
// One wave computes a 16x16 tile of D via WMMA f32. C % 4 == 0 -> no guards.
__global__ void k_dist(const float* __restrict__ x, const float* __restrict__ xx,
                       float* __restrict__ D, int C, int N) {
  int lane = threadIdx.x;
  int half = lane >> 4;
  int l = lane & 15;
  int i0 = blockIdx.y * 16, j0 = blockIdx.x * 16;
  v8f acc = {};
#pragma unroll 4
  for (int c0 = 0; c0 < C; c0 += 4) {
    int c = c0 + 2 * half;  // lane<16: K=c0,c0+1 ; lane>=16: K=c0+2,c0+3
    v2f a, bb;
    a.x = x[c * N + i0 + l];
    a.y = x[(c + 1) * N + i0 + l];
    bb.x = x[c * N + j0 + l];
    bb.y = x[(c + 1) * N + j0 + l];
    acc = wmma4(a, bb, acc);
  }
  float xj = xx[j0 + l];
#pragma unroll
  for (int v = 0; v < 8; v++) {
    int i = i0 + half * 8 + v;  // C/D layout: VGPR v -> M = v + 8*half
    D[(size_t)i * N + j0 + l] = fmaf(-2.f, acc[v], xx[i] + xj);
  }
}

// ---------------------------------------------------------------- top-k (k=20)
// Iterative min-selection; ties -> lowest index (matches jax.lax.top_k on -dist).
// Row staged into LDS with async global->LDS b128 copies (no VGPR round-trip).
__global__ void k_topk(const float* __restrict__ D, int* __restrict__ idx,
                       int N, int k) {
  __shared__ float sd[2048];
  __shared__ float rv[256];
  __shared__ int ri[256];
  int row = blockIdx.x, tid = threadIdx.x;
  const float* dr = D + (size_t)row * N;
  // Each thread async-copies one contiguous 8-float (32B) chunk: 2 x b128.
  {
    const float* g = dr + tid * 8;
    float* lp = &sd[tid * 8];
    async_g2lds_b128(lp, g);
    async_g2lds_b128(lp + 4, g + 4);
    wait_asynccnt0();
  }
  __syncthreads();
  for (int sel = 0; sel < k; sel++) {
    float best = 3.4e38f;
    int bi = N;
    for (int j = tid; j < N; j += 256) {
      float v = sd[j];
      if (v < best) { best = v; bi = j; }  // ascending j keeps lowest tie index
    }
    rv[tid] = best;
    ri[tid] = bi;
    __syncthreads();
    for (int s = 128; s > 0; s >>= 1) {
      if (tid < s) {
        float v2 = rv[tid + s];
        int i2 = ri[tid + s];
        if (v2 < rv[tid] || (v2 == rv[tid] && i2 < ri[tid])) {
          rv[tid] = v2;
          ri[tid] = i2;
        }
      }
      __syncthreads();
    }
    if (tid == 0) {
      idx[(size_t)row * k + sel] = ri[0];
      sd[ri[0]] = 3.4e38f;
    }
    __syncthreads();
  }
}

// --------------------------- EdgeConv: gather + GEMM + BN + lrelu + max over k
// One wave: one point i, one 16-wide output-channel tile. Rows M = k-slots
// (20 valid, padded to 32 = two WMMA row tiles). K-loop split into center-part
// (feature = x_i, identical rows) and diff-part (x_j - x_i): branch-free.
// Invalid padded slots produce garbage rows that are excluded by the max mask.
// grid (O/16, N, B), block 32.
__global__ void k_edgeconv(const float* __restrict__ x, const int* __restrict__ idx,
                           const float* __restrict__ W,
                           const float* __restrict__ scale,
                           const float* __restrict__ shift,
                           float* __restrict__ out, int C, int N, int O, int k) {
  int lane = threadIdx.x;
  int half = lane >> 4;
  int l = lane & 15;
  int b = blockIdx.z;
  int i = blockIdx.y;
  int o0 = blockIdx.x * 16;
  int o = o0 + l;
  const float* xb = x + (size_t)b * C * N;
  const int* ib = idx + (size_t)b * N * k;
  float* ob = out + (size_t)b * O * N;
  int twoC = 2 * C;

  int j1 = ib[(size_t)i * k + l];       // tile1 slot = l (always < k)
  int s2 = 16 + l;                      // tile2 slot; rows >= k masked later
  int j2 = (s2 < k) ? ib[(size_t)i * k + s2] : 0;

  v8f acc1 = {}, acc2 = {};
  // ---- center features: feat[s][c] = x[c, i] for all slots s
#pragma unroll 4
  for (int c0 = 0; c0 < C; c0 += 4) {
    int c = c0 + 2 * half;
    v2f a;
    a.x = xb[c * N + i];
    a.y = xb[(c + 1) * N + i];
    v2f bb = ld2(&W[(size_t)o * twoC + c]);
    acc1 = wmma4(a, bb, acc1);
    acc2 = wmma4(a, bb, acc2);
  }
  // ---- diff features: feat[s][C+c] = x[c, j_s] - x[c, i]
#pragma unroll 4
  for (int c0 = 0; c0 < C; c0 += 4) {
    int c = c0 + 2 * half;
    float cx = xb[c * N + i];
    float cy = xb[(c + 1) * N + i];
    v2f a1, a2;
    a1.x = xb[c * N + j1] - cx;
    a1.y = xb[(c + 1) * N + j1] - cy;
    a2.x = xb[c * N + j2] - cx;
    a2.y = xb[(c + 1) * N + j2] - cy;
    v2f bb = ld2(&W[(size_t)o * twoC + C + c]);
    acc1 = wmma4(a1, bb, acc1);
    acc2 = wmma4(a2, bb, acc2);
  }

  // BN + lrelu then masked max over the k valid rows.
  float sc = scale[o], sh = shift[o];
  float mx = -3.4e38f;
#pragma unroll
  for (int v = 0; v < 8; v++) {  // acc1 rows: slots 0-7 (half0) / 8-15 (half1)
    float y = lrelu(acc1[v] * sc + sh);
    mx = fmaxf(mx, y);
  }
  if (half == 0) {               // acc2 rows 16..23 on half0; valid slots 16..19
#pragma unroll
    for (int v = 0; v < 4; v++) {
      float y = lrelu(acc2[v] * sc + sh);
      mx = fmaxf(mx, y);
    }
  }
  mx = fmaxf(mx, __shfl_xor(mx, 16, 32));  // combine lane halves
  if (half == 0) ob[(size_t)o * N + i] = mx;
}

// ---------------------------------------------------------------- concat x1..x4
__global__ void k_concat(const float* __restrict__ x1, const float* __restrict__ x2,
                         const float* __restrict__ x3, const float* __restrict__ x4,
                         float* __restrict__ xc, int N) {
  int n = blockIdx.x * blockDim.x + threadIdx.x;
  int c = blockIdx.y;
  int b = blockIdx.z;
  if (n >= N) return;
  float v;
  if (c < 64)       v = x1[((size_t)b * 64 + c) * N + n];
  else if (c < 128) v = x2[((size_t)b * 64 + (c - 64)) * N + n];
  else if (c < 256) v = x3[((size_t)b * 128 + (c - 128)) * N + n];
  else              v = x4[((size_t)b * 256 + (c - 256)) * N + n];
  xc[((size_t)b * 512 + c) * N + n] = v;
}

// --------------- emb = lrelu(bn5(W5 @ xc)) fused with per-tile max/sum partials
__global__ void k_w5(const float* __restrict__ xc, const float* __restrict__ W5,
                     const float* __restrict__ scale, const float* __restrict__ shift,
                     float* __restrict__ pmax, float* __restrict__ psum, int N) {
  int lane = threadIdx.x;
  int half = lane >> 4;
  int l = lane & 15;
  int n0 = blockIdx.x * 16, o0 = blockIdx.y * 16, b = blockIdx.z;
  const float* xb = xc + (size_t)b * 512 * N;
  v8f acc = {};
#pragma unroll 4
  for (int c0 = 0; c0 < 512; c0 += 4) {
    int c = c0 + 2 * half;
    v2f a = ld2(&W5[(size_t)(o0 + l) * 512 + c]);
    v2f bb;
    bb.x = xb[(size_t)c * N + n0 + l];
    bb.y = xb[(size_t)(c + 1) * N + n0 + l];
    acc = wmma4(a, bb, acc);
  }
  int nt = N / 16;
#pragma unroll
  for (int v = 0; v < 8; v++) {
    int o = o0 + half * 8 + v;  // row (channel) for this VGPR in this half
    float y = lrelu(acc[v] * scale[o] + shift[o]);
    float mx = y, sm = y;
    for (int d = 1; d < 16; d <<= 1) {  // reduce across the 16 lanes of a half
      mx = fmaxf(mx, __shfl_xor(mx, d, 32));
      sm += __shfl_xor(sm, d, 32);
    }
    if (l == 0) {
      pmax[((size_t)b * 1024 + o) * nt + blockIdx.x] = mx;
      psum[((size_t)b * 1024 + o) * nt + blockIdx.x] = sm;
    }
  }
}

// ---------------------------------------------------------------- final pooling
__global__ void k_pool(const float* __restrict__ pmax, const float* __restrict__ psum,
                       float* __restrict__ feat, int nt, int N) {
  int o = blockIdx.x * blockDim.x + threadIdx.x;
  int b = blockIdx.y;
  if (o >= 1024) return;
  const float* pm = pmax + ((size_t)b * 1024 + o) * nt;
  const float* ps = psum + ((size_t)b * 1024 + o) * nt;
  float mx = -3.4e38f, sm = 0.f;
  for (int t = 0; t < nt; t++) {
    mx = fmaxf(mx, pm[t]);
    sm += ps[t];
  }
  feat[(size_t)b * 2048 + o] = mx;
  feat[(size_t)b * 2048 + 1024 + o] = sm / (float)N;
}

// ------------------- FC: out(BxO) = act(in(BxK) @ W(OxK)^T); M=16 (8 valid rows)
// mode 0: BN(scale,shift)+lrelu. mode 1: +bias (shift) only.
__global__ void k_fc(const float* __restrict__ in, const float* __restrict__ W,
                     const float* __restrict__ scale, const float* __restrict__ shift,
                     float* __restrict__ out, int B, int Kd, int O, int mode) {
  int lane = threadIdx.x;
  int half = lane >> 4;
  int l = lane & 15;
  int o0 = blockIdx.x * 16;
  int o = o0 + l;
  int orow = (o < O) ? o : 0;       // clamp, results for o>=O discarded
  int arow = (l < B) ? l : 0;       // duplicate batch row, rows 8..15 unused
  v8f acc = {};
#pragma unroll 4
  for (int c0 = 0; c0 < Kd; c0 += 4) {
    int c = c0 + 2 * half;
    v2f a = ld2(&in[(size_t)arow * Kd + c]);
    v2f bb = ld2(&W[(size_t)orow * Kd + c]);
    acc = wmma4(a, bb, acc);
  }
  if (half == 0 && o < O) {  // half0 holds rows M=0..7 = all 8 batches
    float sc = (mode == 0) ? scale[o] : 1.f;
    float sh = shift[o];
#pragma unroll
    for (int v = 0; v < 8; v++) {
      float y = acc[v] * sc + sh;
      if (mode == 0) y = lrelu(y);
      if (v < B) out[(size_t)v * O + o] = y;
    }
  }
}

extern "C" void kernel_launch(void* const* d_in, const int* in_sizes, int n_in,
                              void* d_out, int out_size, void* d_ws, size_t ws_size,
                              hipStream_t stream) {
  (void)in_sizes; (void)n_in; (void)out_size; (void)ws_size;
  const int B = 8, N = 2048, K = 20;

  const float* x   = (const float*)d_in[0];
  const float* W1  = (const float*)d_in[1];
  const float* W2  = (const float*)d_in[6];
  const float* W3  = (const float*)d_in[11];
  const float* W4  = (const float*)d_in[16];
  const float* W5  = (const float*)d_in[21];
  const float* L1  = (const float*)d_in[26];
  const float* L2  = (const float*)d_in[31];
  const float* L3  = (const float*)d_in[36];
  const float* L3b = (const float*)d_in[37];

  const int bnbase[7] = {2, 7, 12, 17, 22, 27, 32};
  const int bnC[7] = {64, 64, 128, 256, 1024, 512, 256};

  // Workspace carve (~90 MB total)
  char* w = (char*)d_ws;
  auto alloc = [&](size_t bytes) -> char* {
    char* p = w;
    w += (bytes + 255) & ~(size_t)255;
    return p;
  };
  float* scale[7];
  float* shiftb[7];
  for (int i = 0; i < 7; i++) {
    scale[i] = (float*)alloc((size_t)bnC[i] * 4);
    shiftb[i] = (float*)alloc((size_t)bnC[i] * 4);
  }
  float* xp   = (float*)alloc((size_t)B * 4 * N * 4);        // padded layer-1 input
  float* W1p  = (float*)alloc((size_t)64 * 8 * 4);           // padded W1
  float* xx   = (float*)alloc((size_t)B * N * 4);
  float* Dbuf = (float*)alloc((size_t)N * N * 4);            // 16 MB, per-batch reuse
  int*   idxb = (int*)  alloc((size_t)B * N * K * 4);
  float* x1   = (float*)alloc((size_t)B * 64 * N * 4);
  float* x2   = (float*)alloc((size_t)B * 64 * N * 4);
  float* x3   = (float*)alloc((size_t)B * 128 * N * 4);
  float* x4   = (float*)alloc((size_t)B * 256 * N * 4);
  float* xc   = (float*)alloc((size_t)B * 512 * N * 4);
  float* pmax = (float*)alloc((size_t)B * 1024 * (N / 16) * 4);
  float* psum = (float*)alloc((size_t)B * 1024 * (N / 16) * 4);
  float* feat = (float*)alloc((size_t)B * 2048 * 4);
  float* h1   = (float*)alloc((size_t)B * 512 * 4);
  float* h2   = (float*)alloc((size_t)B * 256 * 4);

  for (int i = 0; i < 7; i++) {
    k_bnprep<<<dim3((bnC[i] + 255) / 256), dim3(256), 0, stream>>>(
        (const float*)d_in[bnbase[i]], (const float*)d_in[bnbase[i] + 1],
        (const float*)d_in[bnbase[i] + 2], (const float*)d_in[bnbase[i] + 3],
        scale[i], shiftb[i], bnC[i]);
  }
  k_pad<<<dim3(N / 256, 4, B), dim3(256), 0, stream>>>(x, xp, N);
  k_padW1<<<dim3(2), dim3(256), 0, stream>>>(W1, W1p);

  struct Layer {
    const float* in;
    const float* W;
    int Cin, Cout;
    float* out;
    int bni;
  } layers[4] = {
      {xp, W1p, 4, 64, x1, 0},
      {x1, W2, 64, 64, x2, 1},
      {x2, W3, 64, 128, x3, 2},
      {x3, W4, 128, 256, x4, 3},
  };

  for (int L = 0; L < 4; L++) {
    Layer& ly = layers[L];
    k_sqnorm<<<dim3((N + 255) / 256, B), dim3(256), 0, stream>>>(ly.in, xx, ly.Cin, N);
    for (int b = 0; b < B; b++) {  // Dbuf holds one batch at a time
      k_dist<<<dim3(N / 16, N / 16), dim3(32), 0, stream>>>(
          ly.in + (size_t)b * ly.Cin * N, xx + (size_t)b * N, Dbuf, ly.Cin, N);
      k_topk<<<dim3(N), dim3(256), 0, stream>>>(Dbuf, idxb + (size_t)b * N * K, N, K);
    }
    k_edgeconv<<<dim3(ly.Cout / 16, N, B), dim3(32), 0, stream>>>(
        ly.in, idxb, ly.W, scale[ly.bni], shiftb[ly.bni], ly.out, ly.Cin, N,
        ly.Cout, K);
  }

  k_concat<<<dim3(N / 256, 512, B), dim3(256), 0, stream>>>(x1, x2, x3, x4, xc, N);
  k_w5<<<dim3(N / 16, 1024 / 16, B), dim3(32), 0, stream>>>(xc, W5, scale[4],
                                                            shiftb[4], pmax, psum, N);
  k_pool<<<dim3(1024 / 256, B), dim3(256), 0, stream>>>(pmax, psum, feat, N / 16, N);

  k_fc<<<dim3(512 / 16), dim3(32), 0, stream>>>(feat, L1, scale[5], shiftb[5], h1,
                                                B, 2048, 512, 0);
  k_fc<<<dim3(256 / 16), dim3(32), 0, stream>>>(h1, L2, scale[6], shiftb[6], h2,
                                                B, 512, 256, 0);
  k_fc<<<dim3((40 + 15) / 16), dim3(32), 0, stream>>>(h2, L3, (const float*)nullptr,
                                                      L3b, (float*)d_out, B, 256, 40, 1);
}